// DualBranchVQC_75883482185867
// MI455X (gfx1250) — compile-verified
//
#include <hip/hip_runtime.h>
#include <hip/hip_bf16.h>
#include <math.h>

typedef __attribute__((ext_vector_type(2))) float v2f;
typedef __attribute__((ext_vector_type(8))) float v8f;

#define PI_F 3.14159265358979323846f

// ---- fast tanh: CDNA5 has V_TANH_F32 transcendental ----
__device__ __forceinline__ float fast_tanh(float x) {
#if __has_builtin(__builtin_amdgcn_tanhf)
  return __builtin_amdgcn_tanhf(x);
#else
  return tanhf(x);
#endif
}

// ---- tiny 16-dim statevector gate helpers (fully unrolled, registers) ----
template<int M>
__device__ __forceinline__ void ry_gate(float s[16], float c, float sn) {
#pragma unroll
  for (int j = 0; j < 16; ++j) {
    if ((j & M) == 0) {
      const int j1 = j | M;
      float a = s[j], b = s[j1];
      s[j]  = c * a - sn * b;
      s[j1] = sn * a + c * b;
    }
  }
}

template<int CM, int TM>
__device__ __forceinline__ void cnot_gate(float s[16]) {
#pragma unroll
  for (int j = 0; j < 16; ++j) {
    if ((j & CM) && !(j & TM)) {
      const int j1 = j | TM;
      float t = s[j]; s[j] = s[j1]; s[j1] = t;
    }
  }
}

__device__ __forceinline__ void ry_layer(float s[16], const float* __restrict__ wrow) {
  float t0 = 0.5f * wrow[0]; ry_gate<8>(s, __cosf(t0), __sinf(t0)); // wire0 -> bit mask 8
  float t1 = 0.5f * wrow[1]; ry_gate<4>(s, __cosf(t1), __sinf(t1));
  float t2 = 0.5f * wrow[2]; ry_gate<2>(s, __cosf(t2), __sinf(t2));
  float t3 = 0.5f * wrow[3]; ry_gate<1>(s, __cosf(t3), __sinf(t3));
}

__global__ __launch_bounds__(256) void DualBranchVQC_75883482185867_kernel(
    const float* __restrict__ sensing, const float* __restrict__ comm,
    const float* __restrict__ weights,
    const float* __restrict__ sp_w, const float* __restrict__ sp_b,
    const float* __restrict__ cp_w, const float* __restrict__ cp_b,
    const float* __restrict__ so_w, const float* __restrict__ so_b,
    const float* __restrict__ co_w, const float* __restrict__ co_b,
    float* __restrict__ out, int nB)
{
  __shared__ float U[256];   // U[j*16 + k] = <j| circuit |k>, fixed per dispatch

  const int tid  = threadIdx.x;
  const int lane = tid & 31;
  const int h    = lane >> 4;   // lane half: selects K-offset (+2) / D-row offset (+8)
  const int n    = lane & 15;   // A-row (M) index AND sample-in-tile / column index

  // ---- build the shared 16x16 circuit matrix U once per block ----
  if (tid < 16) {
    float s[16];
#pragma unroll
    for (int j = 0; j < 16; ++j) s[j] = (j == tid) ? 1.0f : 0.0f;
    ry_layer(s, weights + 0);
    cnot_gate<8, 4>(s);                // CNOT(0,1)
    cnot_gate<2, 1>(s);                // CNOT(2,3)
    ry_layer(s, weights + 4);
    cnot_gate<4, 2>(s);                // CNOT(1,2)
    cnot_gate<1, 8>(s);                // CNOT(3,0)
    ry_layer(s, weights + 8);
    cnot_gate<8, 4>(s);
    cnot_gate<4, 2>(s);
    cnot_gate<2, 1>(s);
    cnot_gate<1, 8>(s);
    ry_layer(s, weights + 12);
#pragma unroll
    for (int j = 0; j < 16; ++j) U[j * 16 + tid] = s[j];
  }
  __syncthreads();

  // ---- stage-1 A operand with PERMUTED rows ----
  // D-layout puts GEMM row m=v+8h in reg v / half h; stage-2's B operand wants
  // circuit row k=4cc+r+2h in reg 2cc+r / half h. Assigning GEMM row m the
  // circuit row pi(m) = 4*((m&7)>>1) + (m&1) + 2*(m>>3) makes the squared
  // stage-1 accumulators directly usable as stage-2 B chunks (no shuffles).
  const int jrow = 4 * ((n & 7) >> 1) + (n & 1) + 2 * (n >> 3);
  v2f a1[4];
#pragma unroll
  for (int cc = 0; cc < 4; ++cc) {
    a1[cc].x = U[jrow * 16 + (4 * cc + 0 + 2 * h)];
    a1[cc].y = U[jrow * 16 + (4 * cc + 1 + 2 * h)];
  }

  // ---- stage-2 A operand: G[m][k] = signed head weights (rows m=0..2) ----
  // Columns k are circuit basis indices, matching the B-layout K mapping above.
  float g2w0 = 0.0f, g2w1 = 0.0f; int ma = 8, mb = 4;
  if (n == 0)      { g2w0 = so_w[0]; g2w1 = so_w[1]; }           // sens ch0 on ez0,ez1
  else if (n == 1) { g2w0 = so_w[2]; g2w1 = so_w[3]; }           // sens ch1 on ez0,ez1
  else if (n == 2) { g2w0 = co_w[0]; g2w1 = co_w[1]; ma = 2; mb = 1; } // comm on ez2,ez3
  v2f a2[4];
#pragma unroll
  for (int cc = 0; cc < 4; ++cc) {
    const int k0 = 4 * cc + 2 * h, k1 = k0 + 1;
    a2[cc].x = g2w0 * ((k0 & ma) ? -1.0f : 1.0f) + g2w1 * ((k0 & mb) ? -1.0f : 1.0f);
    a2[cc].y = g2w0 * ((k1 & ma) ? -1.0f : 1.0f) + g2w1 * ((k1 & mb) ? -1.0f : 1.0f);
  }

  // ---- stage-2 C operand: bias in rows 0..2 (lanes 0-15 hold rows 0..7) ----
  v8f cbias = {0.f, 0.f, 0.f, 0.f, 0.f, 0.f, 0.f, 0.f};
  if (h == 0) { cbias[0] = so_b[0]; cbias[1] = so_b[1]; cbias[2] = co_b[0]; }

  // ---- uniform projection params ----
  const float spw00 = sp_w[0], spw01 = sp_w[1], spw10 = sp_w[2], spw11 = sp_w[3];
  const float spb0 = sp_b[0], spb1 = sp_b[1];
  const float cpw0 = cp_w[0], cpw1 = cp_w[1], cpb0 = cp_b[0], cpb1 = cp_b[1];
  const float halfpi = 0.5f * PI_F;

  const int wavesPerBlock = blockDim.x >> 5;
  const int waveId = blockIdx.x * wavesPerBlock + (tid >> 5);
  const int nWaves = gridDim.x * wavesPerBlock;
  const int nTiles = (nB + 15) >> 4;
  const float2* __restrict__ sens2 = reinterpret_cast<const float2*>(sensing);

  for (int t = waveId; t < nTiles; t += nWaves) {
    const int idx = t * 16 + n;
    const int si  = (idx < nB) ? idx : (nB - 1);

    // prefetch next tile for this wave (speculative; dropped if invalid)
    __builtin_prefetch(&sens2[(size_t)si + (size_t)16 * nWaves], 0, 1);
    __builtin_prefetch(&comm [(size_t)si + (size_t)16 * nWaves], 0, 1);

    const float2 sv = sens2[si];
    const float  cmv = comm[si];

    // half-angles of the 4 input rotations
    const float ha0 = fast_tanh(spw00 * sv.x + spw01 * sv.y + spb0) * halfpi;
    const float ha1 = fast_tanh(spw10 * sv.x + spw11 * sv.y + spb1) * halfpi;
    const float ha2 = fast_tanh(cpw0 * cmv + cpb0) * halfpi;
    const float ha3 = fast_tanh(cpw1 * cmv + cpb1) * halfpi;
    const float c0 = __cosf(ha0), s0 = __sinf(ha0);
    const float c1 = __cosf(ha1), s1 = __sinf(ha1);
    const float c2 = __cosf(ha2), s2 = __sinf(ha2);
    const float c3 = __cosf(ha3), s3 = __sinf(ha3);

    // st0[k] = f0(q0)*f1(q1)*f2(q2)*f3(q3); this lane needs k = 4*cc + r + 2*h
    // => (q0,q1)=bits(cc), q2=h, q3=r
    const float f2h = h ? s2 : c2;
    const float pc = f2h * c3, ps = f2h * s3;
    const float g0 = c0 * c1, g1 = c0 * s1, g2v = s0 * c1, g3 = s0 * s1;
    v2f b1[4];
    b1[0].x = g0 * pc;  b1[0].y = g0 * ps;
    b1[1].x = g1 * pc;  b1[1].y = g1 * ps;
    b1[2].x = g2v * pc; b1[2].y = g2v * ps;
    b1[3].x = g3 * pc;  b1[3].y = g3 * ps;

    // ---- stage 1: D[pi(m)][sample] = U(permuted rows) x st0^T ----
    v8f d = {0.f, 0.f, 0.f, 0.f, 0.f, 0.f, 0.f, 0.f};
#pragma unroll
    for (int cc = 0; cc < 4; ++cc)
      d = __builtin_amdgcn_wmma_f32_16x16x4_f32(false, a1[cc], false, b1[cc],
                                                (short)0, d, false, false);

    // probabilities: squared accumulators ARE the stage-2 B chunks (pairwise)
    v8f p;
#pragma unroll
    for (int v = 0; v < 8; ++v) p[v] = d[v] * d[v];

    // ---- stage 2: out3 = G x p + bias ----
    v8f d2 = cbias;
#pragma unroll
    for (int cc = 0; cc < 4; ++cc) {
      v2f b2; b2.x = p[2 * cc]; b2.y = p[2 * cc + 1];
      d2 = __builtin_amdgcn_wmma_f32_16x16x4_f32(false, a2[cc], false, b2,
                                                 (short)0, d2, false, false);
    }

    // rows 0,1 = sensing_final, row 2 = comm_final; columns = samples (lanes 0-15)
    if (h == 0 && idx < nB) {
      float2 sf; sf.x = d2[0]; sf.y = d2[1];
      *reinterpret_cast<float2*>(out + 2 * (size_t)idx) = sf;
      out[2 * (size_t)nB + idx] = d2[2];
    }
  }
}

extern "C" void kernel_launch(void* const* d_in, const int* in_sizes, int n_in,
                              void* d_out, int out_size, void* d_ws, size_t ws_size,
                              hipStream_t stream) {
  const float* sensing = (const float*)d_in[0];
  const float* comm    = (const float*)d_in[1];
  const float* weights = (const float*)d_in[2];
  const float* sp_w    = (const float*)d_in[3];
  const float* sp_b    = (const float*)d_in[4];
  const float* cp_w    = (const float*)d_in[5];
  const float* cp_b    = (const float*)d_in[6];
  const float* so_w    = (const float*)d_in[7];
  const float* so_b    = (const float*)d_in[8];
  const float* co_w    = (const float*)d_in[9];
  const float* co_b    = (const float*)d_in[10];
  float* out = (float*)d_out;

  const int nB = in_sizes[1];            // comm has B elements
  const int threads = 256;               // 8 wave32s per block
  const int wavesPerBlock = threads / 32;
  const int nTiles = (nB + 15) / 16;
  int blocks = 2048;
  const int maxBlocks = (nTiles + wavesPerBlock - 1) / wavesPerBlock;
  if (blocks > maxBlocks) blocks = maxBlocks;
  if (blocks < 1) blocks = 1;

  DualBranchVQC_75883482185867_kernel<<<dim3(blocks), dim3(threads), 0, stream>>>(
      sensing, comm, weights, sp_w, sp_b, cp_w, cp_b, so_w, so_b, co_w, co_b,
      out, nB);
}